// GraphSageLayer_49082886258797
// MI455X (gfx1250) — compile-verified
//
#include <hip/hip_runtime.h>

typedef float v2f __attribute__((ext_vector_type(2)));
typedef float v4f __attribute__((ext_vector_type(4)));
typedef float v8f __attribute__((ext_vector_type(8)));

#define N_NODES 50000
#define N_EDGES 800000
#define D_IN    64
#define D_OUT   64
#define D2      128   // 2*D_IN

// ---------------------------------------------------------------------------
// Kernel 1: zero the neighbor accumulator (N_NODES * D_IN floats in ws)
// 3,200,000 floats = 800,000 float4; 256 threads * 3125 blocks exactly.
// ---------------------------------------------------------------------------
__global__ void gsage_zero_ws(float* __restrict__ ws) {
    int i = blockIdx.x * blockDim.x + threadIdx.x;
    ((v4f*)ws)[i] = (v4f){0.f, 0.f, 0.f, 0.f};
}

// ---------------------------------------------------------------------------
// Kernel 2: edge scatter: nbr[dst] += features[src] * w  (atomic f32 adds)
// One wave (32 lanes) per edge; each lane handles 2 of the 64 dims.
// 800,000 edges = 100,000 blocks * 8 waves exactly -> no guard needed.
// ---------------------------------------------------------------------------
__global__ void gsage_edge_scatter(const float* __restrict__ feat,
                                   const int*   __restrict__ esrc,
                                   const int*   __restrict__ edst,
                                   const float* __restrict__ ew,
                                   float*       __restrict__ nbr) {
    int tid  = blockIdx.x * blockDim.x + threadIdx.x;
    int e    = tid >> 5;
    int lane = tid & 31;
    int s  = esrc[e];
    int d  = edst[e];
    float w = ew[e];
    int c = lane * 2;
    v2f f = *(const v2f*)(feat + (size_t)s * D_IN + c);
    atomicAdd(nbr + (size_t)d * D_IN + c,     f.x * w);
    atomicAdd(nbr + (size_t)d * D_IN + c + 1, f.y * w);
}

// ---------------------------------------------------------------------------
// Kernel 3: out = normalize( [feat | nbr] @ W^T + b ) using V_WMMA_F32_16X16X4_F32
//
// One wave per 16-row tile; 4 n-tiles (64 output cols) per wave.
// 50000/16 = 3125 tiles = 625 blocks * 5 waves exactly -> EXEC all-1s always.
//
// f32 16x16x4 fragment layout (ISA 7.12.2):
//   A (16x4): lanes 0-15 -> M=lane, v0=K+0, v1=K+1 ; lanes 16-31 -> v0=K+2, v1=K+3
//   B (4x16): lanes 0-15 -> N=lane, v0=K+0, v1=K+1 ; lanes 16-31 -> v0=K+2, v1=K+3
//   C/D (16x16): lanes 0-15 -> N=lane, VGPR v -> M=v ; lanes 16-31 -> M=v+8
// ---------------------------------------------------------------------------
__global__ void gsage_gemm_norm(const float* __restrict__ feat,
                                const float* __restrict__ nbr,
                                const float* __restrict__ W,
                                const float* __restrict__ bias,
                                float*       __restrict__ out) {
    const int wavesPerBlock = blockDim.x >> 5;
    const int wave = threadIdx.x >> 5;
    const int lane = threadIdx.x & 31;
    const int tile = blockIdx.x * wavesPerBlock + wave;  // < 3125 by construction

    const int r0   = tile * 16;
    const int mrow = lane & 15;       // M (for A) or N (for B/C/D) position
    const int half = lane >> 4;       // 0: K+{0,1}, 1: K+{2,3}
    const int koff = half * 2;

    v8f acc0 = {}, acc1 = {}, acc2 = {}, acc3 = {};

    // A row pointers (K<64 from features, K>=64 from neighbor accumulator)
    const float* __restrict__ a_feat = feat + (size_t)(r0 + mrow) * D_IN + koff;
    const float* __restrict__ a_nbr  = nbr  + (size_t)(r0 + mrow) * D_IN + koff;
    // B: B[kk][n] = W[n*128 + kk]; per-lane column base
    const float* __restrict__ w0 = W + (size_t)( 0 + mrow) * D2 + koff;
    const float* __restrict__ w1 = W + (size_t)(16 + mrow) * D2 + koff;
    const float* __restrict__ w2 = W + (size_t)(32 + mrow) * D2 + koff;
    const float* __restrict__ w3 = W + (size_t)(48 + mrow) * D2 + koff;

    // ---- K = 0..63 (features half) ----
    #pragma unroll 4
    for (int k = 0; k < D_IN; k += 4) {
        v2f a  = *(const v2f*)(a_feat + k);
        v2f b0 = *(const v2f*)(w0 + k);
        v2f b1 = *(const v2f*)(w1 + k);
        v2f b2 = *(const v2f*)(w2 + k);
        v2f b3 = *(const v2f*)(w3 + k);
        acc0 = __builtin_amdgcn_wmma_f32_16x16x4_f32(false, a, false, b0, (short)0, acc0, false, false);
        acc1 = __builtin_amdgcn_wmma_f32_16x16x4_f32(false, a, false, b1, (short)0, acc1, false, false);
        acc2 = __builtin_amdgcn_wmma_f32_16x16x4_f32(false, a, false, b2, (short)0, acc2, false, false);
        acc3 = __builtin_amdgcn_wmma_f32_16x16x4_f32(false, a, false, b3, (short)0, acc3, false, false);
    }
    // ---- K = 64..127 (neighbor half) ----
    #pragma unroll 4
    for (int k = 0; k < D_IN; k += 4) {
        v2f a  = *(const v2f*)(a_nbr + k);
        v2f b0 = *(const v2f*)(w0 + D_IN + k);
        v2f b1 = *(const v2f*)(w1 + D_IN + k);
        v2f b2 = *(const v2f*)(w2 + D_IN + k);
        v2f b3 = *(const v2f*)(w3 + D_IN + k);
        acc0 = __builtin_amdgcn_wmma_f32_16x16x4_f32(false, a, false, b0, (short)0, acc0, false, false);
        acc1 = __builtin_amdgcn_wmma_f32_16x16x4_f32(false, a, false, b1, (short)0, acc1, false, false);
        acc2 = __builtin_amdgcn_wmma_f32_16x16x4_f32(false, a, false, b2, (short)0, acc2, false, false);
        acc3 = __builtin_amdgcn_wmma_f32_16x16x4_f32(false, a, false, b3, (short)0, acc3, false, false);
    }

    // ---- epilogue: bias, L2 row norm, normalize, store ----
    const float bb0 = bias[ 0 + mrow];
    const float bb1 = bias[16 + mrow];
    const float bb2 = bias[32 + mrow];
    const float bb3 = bias[48 + mrow];

    float ss[8];
    #pragma unroll
    for (int v = 0; v < 8; ++v) {
        acc0[v] += bb0; acc1[v] += bb1; acc2[v] += bb2; acc3[v] += bb3;
        ss[v] = acc0[v]*acc0[v] + acc1[v]*acc1[v] + acc2[v]*acc2[v] + acc3[v]*acc3[v];
    }
    // Reduce across the 16 lanes of each half-wave (masks < 16 keep halves separate).
    #pragma unroll
    for (int m = 1; m < 16; m <<= 1) {
        #pragma unroll
        for (int v = 0; v < 8; ++v) ss[v] += __shfl_xor(ss[v], m, 32);
    }
    // Lane (half,*) VGPR v holds row (r0 + v + half*8), col (16*t + mrow).
    #pragma unroll
    for (int v = 0; v < 8; ++v) {
        const float inv = 1.0f / fmaxf(sqrtf(ss[v]), 1e-12f);
        const size_t row = (size_t)(r0 + v + half * 8) * D_OUT;
        out[row +  0 + mrow] = acc0[v] * inv;
        out[row + 16 + mrow] = acc1[v] * inv;
        out[row + 32 + mrow] = acc2[v] * inv;
        out[row + 48 + mrow] = acc3[v] * inv;
    }
}

// ---------------------------------------------------------------------------
extern "C" void kernel_launch(void* const* d_in, const int* in_sizes, int n_in,
                              void* d_out, int out_size, void* d_ws, size_t ws_size,
                              hipStream_t stream) {
    const float* feat = (const float*)d_in[0];
    const int*   esrc = (const int*)  d_in[1];
    const int*   edst = (const int*)  d_in[2];
    const float* ew   = (const float*)d_in[3];
    const float* W    = (const float*)d_in[4];
    const float* bias = (const float*)d_in[5];
    float*       out  = (float*)d_out;
    float*       nbr  = (float*)d_ws;   // N_NODES * D_IN floats (12.8 MB)

    // 1) zero neighbor accumulator: 800,000 float4 / 256 threads = 3125 blocks
    gsage_zero_ws<<<3125, 256, 0, stream>>>(nbr);

    // 2) edge scatter: 800,000 waves / 8 waves-per-block = 100,000 blocks
    gsage_edge_scatter<<<100000, 256, 0, stream>>>(feat, esrc, edst, ew, nbr);

    // 3) fused GEMM + bias + L2-normalize: 3125 tiles = 625 blocks * 5 waves
    gsage_gemm_norm<<<625, 160, 0, stream>>>(feat, nbr, W, bias, out);
}